// SAGPool_71717363908784
// MI455X (gfx1250) — compile-verified
//
#include <hip/hip_runtime.h>
#include <math.h>

typedef __attribute__((ext_vector_type(2))) float v2f;
typedef __attribute__((ext_vector_type(8))) float v8f;

#define NGRAPHS   128
#define NODESPER  1024
#define DEGC      32
#define DFEAT     256
#define NTOT      (NGRAPHS * NODESPER)        // 131072
#define ETOT      (NTOT * DEGC)               // 4194304
#define KKEEP     820
#define GK        (NGRAPHS * KKEEP)           // 104960

// ---------------- init: deg = 1 (self loop), agg = 0 ----------------
__global__ void k_init(float* __restrict__ deg, float* __restrict__ agg) {
  int i = blockIdx.x * blockDim.x + threadIdx.x;
  if (i < NTOT) { deg[i] = 1.0f; agg[i] = 0.0f; }
}

// ---------------- degree accumulate at targets ----------------
__global__ void k_deg(const int* __restrict__ ei, float* __restrict__ deg) {
  int e = blockIdx.x * blockDim.x + threadIdx.x;
  if (e < ETOT) atomicAdd(&deg[ei[ETOT + e]], 1.0f);
}

// ---------------- dinv = rsqrt(deg), in place ----------------
__global__ void k_dinv(float* __restrict__ deg) {
  int i = blockIdx.x * blockDim.x + threadIdx.x;
  if (i < NTOT) deg[i] = rsqrtf(deg[i]);
}

// ---------------- h = x @ W via V_WMMA_F32_16X16X4_F32 ----------------
// One wave computes 16 node scores. A = x tile (16 x 4 per step),
// B = W segment broadcast into all 16 N-columns, so D[m][n] == h[m] for all n.
// A layout (32-bit 16x4): lanes 0-15 hold K={0,1}, lanes 16-31 hold K={2,3}.
// C/D layout: lanes 0-15 VGPR r = M=r; lanes 16-31 VGPR r = M=8+r.
__global__ void k_hscore_wmma(const float* __restrict__ x,
                              const float* __restrict__ W,
                              float* __restrict__ h) {
  int wave = (blockIdx.x * blockDim.x + threadIdx.x) >> 5;
  int lane = threadIdx.x & 31;
  int base = wave * 16;                 // grid sized so base < NTOT always
  int m    = base + (lane & 15);
  int khalf = (lane >> 4) << 1;         // 0 for lanes 0-15, 2 for lanes 16-31
  const float* xrow = x + (size_t)m * DFEAT;

  v8f c = {};
  for (int k = 0; k < DFEAT; k += 4) {
    v2f a; a.x = xrow[k + khalf];     a.y = xrow[k + khalf + 1];
    v2f b; b.x = W[k + khalf];        b.y = W[k + khalf + 1];
    // D = A*B + C   (8 args: neg_a, A, neg_b, B, c_mod, C, reuse_a, reuse_b)
    c = __builtin_amdgcn_wmma_f32_16x16x4_f32(false, a, false, b,
                                              (short)0, c, false, false);
  }
  if (lane == 0) {
    #pragma unroll
    for (int r = 0; r < 8; ++r) h[base + r] = c[r];
  } else if (lane == 16) {
    #pragma unroll
    for (int r = 0; r < 8; ++r) h[base + 8 + r] = c[r];
  }
}

// ---------------- edge aggregation: agg[c] += dinv[r]*dinv[c]*h[r] ----------------
__global__ void k_agg(const int* __restrict__ ei, const float* __restrict__ dinv,
                      const float* __restrict__ h, float* __restrict__ agg) {
  int e = blockIdx.x * blockDim.x + threadIdx.x;
  if (e < ETOT) {
    int r = ei[e];
    int c = ei[ETOT + e];
    atomicAdd(&agg[c], dinv[r] * dinv[c] * h[r]);
  }
}

// ---------------- score = agg + dinv^2 * h + b ----------------
__global__ void k_score(const float* __restrict__ agg, const float* __restrict__ dinv,
                        const float* __restrict__ h, const float* __restrict__ bptr,
                        float* __restrict__ score) {
  int i = blockIdx.x * blockDim.x + threadIdx.x;
  if (i < NTOT) score[i] = agg[i] + dinv[i] * dinv[i] * h[i] + bptr[0];
}

// ---------------- per-graph top-K via LDS bitonic sort ----------------
// One 1024-thread block per graph; sort (score desc, idx asc); keep first K.
__global__ __launch_bounds__(NODESPER) void
k_topk(const float* __restrict__ score, int* __restrict__ new_id,
       int* __restrict__ perm, float* __restrict__ out_batch,
       float* __restrict__ out_perm) {
  __shared__ float s[NODESPER];
  __shared__ int   id[NODESPER];
  int g = blockIdx.x, t = threadIdx.x;

  s[t]  = score[g * NODESPER + t];
  id[t] = t;
  new_id[g * NODESPER + t] = -1;
  __syncthreads();

  for (int k = 2; k <= NODESPER; k <<= 1) {
    for (int j = k >> 1; j > 0; j >>= 1) {
      int ixj = t ^ j;
      if (ixj > t) {
        float s0 = s[t], s1 = s[ixj];
        int   i0 = id[t], i1 = id[ixj];
        // "t after ixj" in descending-with-index-tiebreak order
        bool bad = (s0 < s1) || (s0 == s1 && i0 > i1);
        if (bad == ((t & k) == 0)) {
          s[t] = s1; s[ixj] = s0;
          id[t] = i1; id[ixj] = i0;
        }
      }
      __syncthreads();
    }
  }

  if (t < KKEEP) {
    int p  = g * NODESPER + id[t];
    int np = g * KKEEP + t;
    perm[np]      = p;
    new_id[p]     = np;
    out_perm[np]  = (float)p;
    out_batch[np] = (float)g;
  }
}

// ---------------- x_new = x[perm] * tanh(score[perm]) ----------------
__global__ void k_gather(const float* __restrict__ x, const float* __restrict__ score,
                         const int* __restrict__ perm, float* __restrict__ xnew) {
  int idx = blockIdx.x * blockDim.x + threadIdx.x;   // over GK * (D/4)
  const int c4n = DFEAT / 4;
  int row = idx / c4n, c4 = idx % c4n;
  if (row < GK) {
    int p = perm[row];
    float sc = tanhf(score[p]);
    float4 v = ((const float4*)(x + (size_t)p * DFEAT))[c4];
    v.x *= sc; v.y *= sc; v.z *= sc; v.w *= sc;
    ((float4*)(xnew + (size_t)row * DFEAT))[c4] = v;
  }
}

// ---------------- edge remap + mask ----------------
__global__ void k_edges(const int* __restrict__ ei, const int* __restrict__ new_id,
                        float* __restrict__ out_ei, float* __restrict__ out_mask) {
  int e = blockIdx.x * blockDim.x + threadIdx.x;
  if (e < ETOT) {
    int r2 = new_id[ei[e]];
    int c2 = new_id[ei[ETOT + e]];
    bool m = (r2 >= 0) && (c2 >= 0);
    out_ei[e]        = (float)(m ? r2 : -1);
    out_ei[ETOT + e] = (float)(m ? c2 : -1);
    out_mask[e]      = m ? 1.0f : 0.0f;
  }
}

extern "C" void kernel_launch(void* const* d_in, const int* in_sizes, int n_in,
                              void* d_out, int out_size, void* d_ws, size_t ws_size,
                              hipStream_t stream) {
  const float* x   = (const float*)d_in[0];
  const int*   ei  = (const int*)d_in[1];    // [2, E] flat
  // d_in[2] = batch (unused; batch_new derivable from graph id)
  const float* W   = (const float*)d_in[3];
  const float* b   = (const float*)d_in[4];

  // workspace carve (all 4-byte elements)
  float* deg   = (float*)d_ws;               // N  (becomes dinv in place)
  float* agg   = deg + NTOT;                 // N
  float* h     = agg + NTOT;                 // N
  float* score = h + NTOT;                   // N
  int*   newid = (int*)(score + NTOT);       // N
  int*   perm  = newid + NTOT;               // GK

  // output carve (floats, reference return order)
  float* out_x     = (float*)d_out;          // GK*D
  float* out_ei    = out_x + (size_t)GK * DFEAT;  // 2E
  float* out_mask  = out_ei + 2 * (size_t)ETOT;   // E
  float* out_batch = out_mask + (size_t)ETOT;     // GK
  float* out_perm  = out_batch + GK;              // GK

  const int T = 256;
  k_init <<<(NTOT + T - 1) / T, T, 0, stream>>>(deg, agg);
  k_deg  <<<(ETOT + T - 1) / T, T, 0, stream>>>(ei, deg);
  k_dinv <<<(NTOT + T - 1) / T, T, 0, stream>>>(deg);

  // 16 nodes per wave, 8 waves per 256-thread block -> NTOT/128 blocks (exact)
  k_hscore_wmma<<<NTOT / 128, T, 0, stream>>>(x, W, h);

  k_agg  <<<(ETOT + T - 1) / T, T, 0, stream>>>(ei, deg, h, agg);
  k_score<<<(NTOT + T - 1) / T, T, 0, stream>>>(agg, deg, h, b, score);

  k_topk <<<NGRAPHS, NODESPER, 0, stream>>>(score, newid, perm, out_batch, out_perm);

  const int gtot = GK * (DFEAT / 4);
  k_gather<<<(gtot + T - 1) / T, T, 0, stream>>>(x, score, perm, out_x);
  k_edges <<<(ETOT + T - 1) / T, T, 0, stream>>>(ei, newid, out_ei, out_mask);
}